// SGC_609885356314
// MI455X (gfx1250) — compile-verified
//
#include <hip/hip_runtime.h>
#include <hip/hip_bf16.h>

typedef float v2f __attribute__((ext_vector_type(2)));
typedef float v8f __attribute__((ext_vector_type(8)));

#define FEAT 64

// ---------------- degree / normalization ----------------

__global__ void sgc_deg_init(float* __restrict__ deg, int n) {
    int i = blockIdx.x * blockDim.x + threadIdx.x;
    if (i < n) deg[i] = 1.0f;   // self-loop contributes 1
}

__global__ void sgc_deg_accum(const int* __restrict__ dst_idx, float* __restrict__ deg, int nE) {
    int e = blockIdx.x * blockDim.x + threadIdx.x;
    if (e < nE) atomicAdd(&deg[dst_idx[e]], 1.0f);
}

__global__ void sgc_dinv(float* __restrict__ d, int n) {
    int i = blockIdx.x * blockDim.x + threadIdx.x;
    if (i < n) d[i] = rsqrtf(d[i]);   // deg >= 1 always (self loops)
}

// ---------------- SpMM hop: self-loop init + edge scatter ----------------

// h_out[i][:] = dinv[i]^2 * h_in[i][:]   (one thread per float2 of features)
__global__ void sgc_self_init(const float* __restrict__ h_in, const float* __restrict__ dinv,
                              float* __restrict__ h_out, int n) {
    int t = blockIdx.x * blockDim.x + threadIdx.x;
    if (t >= n * (FEAT / 2)) return;
    int node = t >> 5;          // FEAT/2 == 32 pairs per node
    int pair = t & 31;
    float dv = dinv[node];
    float w = dv * dv;
    v2f val = *((const v2f*)(h_in + (size_t)node * FEAT) + pair);
    v2f r; r.x = val.x * w; r.y = val.y * w;
    *((v2f*)(h_out + (size_t)node * FEAT) + pair) = r;
}

// One wave32 per edge; lane handles features [2*lane, 2*lane+1].
__global__ void sgc_edge_scatter(const int* __restrict__ src_idx, const int* __restrict__ dst_idx,
                                 const float* __restrict__ dinv, const float* __restrict__ h_in,
                                 float* __restrict__ h_out, int nE) {
    int t = blockIdx.x * blockDim.x + threadIdx.x;
    int e = t >> 5;
    if (e >= nE) return;
    int pair = t & 31;
    int s = src_idx[e];          // wave-uniform load (broadcast from cache)
    int d = dst_idx[e];
    float w = dinv[s] * dinv[d];
    v2f val = *((const v2f*)(h_in + (size_t)s * FEAT) + pair);
    float* outp = h_out + (size_t)d * FEAT + pair * 2;
    atomicAdd(outp,     val.x * w);
    atomicAdd(outp + 1, val.y * w);
}

// ---------------- final linear: out = h @ W + b  (WMMA f32 16x16x4) ----------------
// One wave computes a 16-row x 64-col output stripe.
// W is staged in LDS pre-packed so each lane's B operand for a k-step is one
// contiguous 64-bit LDS load:  Wp[(kpair*64 + ncol)*2 + j] = W[(2*kpair+j)*64 + ncol]
__global__ void __launch_bounds__(256)
sgc_gemm_wmma(const float* __restrict__ h, const float* __restrict__ W,
              const float* __restrict__ bias, float* __restrict__ out, int n_rows) {
    __shared__ float Ws[FEAT * FEAT];   // 16 KB, packed layout
    for (int i = threadIdx.x; i < FEAT * FEAT; i += blockDim.x) {
        int p    = i >> 7;        // k-pair index (32 pairs)
        int rem  = i & 127;
        int ncol = rem >> 1;
        int j    = rem & 1;
        Ws[i] = W[(2 * p + j) * FEAT + ncol];
    }
    __syncthreads();

    int wave = threadIdx.x >> 5;
    int lane = threadIdx.x & 31;
    int m_base = (blockIdx.x * (blockDim.x >> 5) + wave) * 16;
    if (m_base >= n_rows) return;     // wave-uniform: EXEC stays all-1s for WMMA

    int half = lane >> 4;             // 0: K=0,1   1: K=2,3  (A / B striping)
    int lm   = lane & 15;

    // A layout for V_WMMA_F32_16X16X4_F32: lane<16 -> M=lane, VGPR0/1 = K 0/1;
    // lane>=16 -> M=lane-16, VGPR0/1 = K 2/3. Preload full 16x64 A tile (16 k-steps).
    int row = m_base + lm;
    if (row >= n_rows) row = n_rows - 1;       // safety (N % 16 == 0 in practice)
    const float* hrow = h + (size_t)row * FEAT;
    v2f a[16];
#pragma unroll
    for (int ks = 0; ks < 16; ++ks)
        a[ks] = *(const v2f*)(hrow + ks * 4 + half * 2);

#pragma unroll
    for (int nt = 0; nt < 4; ++nt) {
        int ncol = nt * 16 + lm;
        v8f c = {};
#pragma unroll
        for (int ks = 0; ks < 16; ++ks) {
            int p = ks * 2 + half;             // k-pair this half consumes (K+0/1 or K+2/3)
            v2f b = *(const v2f*)(Ws + ((size_t)p * FEAT + ncol) * 2);  // one ds_load_b64
            c = __builtin_amdgcn_wmma_f32_16x16x4_f32(
                    /*neg_a=*/false, a[ks], /*neg_b=*/false, b,
                    /*c_mod=*/(short)0, c, /*reuse_a=*/false, /*reuse_b=*/false);
        }
        float bv = bias[ncol];                 // same N for all 8 C VGPRs in a lane
#pragma unroll
        for (int v = 0; v < 8; ++v) {
            int m = m_base + v + half * 8;     // C VGPR v: M=v (lanes 0-15), M=v+8 (lanes 16-31)
            out[(size_t)m * FEAT + ncol] = c[v] + bv;
        }
    }
}

// ---------------- launch ----------------

extern "C" void kernel_launch(void* const* d_in, const int* in_sizes, int n_in,
                              void* d_out, int out_size, void* d_ws, size_t ws_size,
                              hipStream_t stream) {
    const float* x    = (const float*)d_in[0];   // [N,64]
    const int*   ei   = (const int*)d_in[1];     // [2,E]
    const float* W    = (const float*)d_in[2];   // [64,64]
    const float* bias = (const float*)d_in[3];   // [64]
    float*       out  = (float*)d_out;           // [N,64]

    const int N = in_sizes[0] / FEAT;
    const int E = in_sizes[1] / 2;
    const int* src = ei;        // edge_index[0]
    const int* dst = ei + E;    // edge_index[1]

    // workspace layout: dinv [N] | h1 [N*64] | h2 [N*64]
    float* dinv = (float*)d_ws;
    size_t nAlign = ((size_t)N + 255) & ~(size_t)255;
    float* h1 = dinv + nAlign;
    float* h2 = h1 + (size_t)N * FEAT;

    const int B = 256;
    int gN  = (N + B - 1) / B;
    int gE  = (E + B - 1) / B;
    int gNF = (int)(((size_t)N * (FEAT / 2) + B - 1) / B);
    int gEW = (int)(((size_t)E * 32 + B - 1) / B);

    // degree (A + I), then dinv = rsqrt(deg)
    sgc_deg_init<<<gN, B, 0, stream>>>(dinv, N);
    sgc_deg_accum<<<gE, B, 0, stream>>>(dst, dinv, E);
    sgc_dinv<<<gN, B, 0, stream>>>(dinv, N);

    // hop 1: x -> h1
    sgc_self_init<<<gNF, B, 0, stream>>>(x, dinv, h1, N);
    sgc_edge_scatter<<<gEW, B, 0, stream>>>(src, dst, dinv, x, h1, E);
    // hop 2: h1 -> h2
    sgc_self_init<<<gNF, B, 0, stream>>>(h1, dinv, h2, N);
    sgc_edge_scatter<<<gEW, B, 0, stream>>>(src, dst, dinv, h1, h2, E);

    // out = h2 @ W + b  (WMMA f32)
    int tiles = (N + 15) / 16;
    int gG = (tiles + (B / 32) - 1) / (B / 32);
    sgc_gemm_wmma<<<gG, B, 0, stream>>>(h2, W, bias, out, N);
}